// AttentionGNN_17970143166729
// MI455X (gfx1250) — compile-verified
//
#include <hip/hip_runtime.h>

// TGCN (GCN + GRU over T=12 steps) restructured for MI455X (gfx1250):
//  - aggregate raw features once (192 floats/edge instead of 1152)
//  - CSR build on device -> gather-based aggregation (no fp32 atomic storm)
//  - GRU dense math on V_WMMA_F32_16X16X4_F32 (exact fp32 tensor-core path)

#define N_NODES   50000
#define F_INF     16
#define T_STEPS   12
#define N_EDGES   1600000
#define HID       32
#define NODE_F    192          // F_INF * T_STEPS

typedef float v2f __attribute__((ext_vector_type(2)));
typedef float v8f __attribute__((ext_vector_type(8)));

// ---------------------------------------------------------------- utilities
__global__ void zero_i32(int* p, int n) {
    int i = blockIdx.x * blockDim.x + threadIdx.x;
    if (i < n) p[i] = 0;
}

__global__ void count_indeg(const int* __restrict__ dst, int* __restrict__ indeg, int e) {
    int i = blockIdx.x * blockDim.x + threadIdx.x;
    if (i < e) atomicAdd(&indeg[dst[i]], 1);
}

__global__ void compute_dis(const int* __restrict__ indeg, float* __restrict__ dis, int n) {
    int i = blockIdx.x * blockDim.x + threadIdx.x;
    if (i < n) dis[i] = rsqrtf((float)(indeg[i] + 1));   // +1 = self loop, always > 0
}

// single-block serial-chunk scan over N=50000 (tiny vs rest of pipeline)
__global__ void scan_offsets(const int* __restrict__ indeg, int* __restrict__ offs,
                             int* __restrict__ cursor, int n) {
    __shared__ int buf[1024];
    __shared__ int carry;
    int tid = threadIdx.x;
    if (tid == 0) carry = 0;
    __syncthreads();
    for (int base = 0; base < n; base += 1024) {
        int v = (base + tid < n) ? indeg[base + tid] : 0;
        buf[tid] = v;
        __syncthreads();
        for (int off = 1; off < 1024; off <<= 1) {
            int t = (tid >= off) ? buf[tid - off] : 0;
            __syncthreads();
            buf[tid] += t;
            __syncthreads();
        }
        int excl = carry + buf[tid] - v;
        if (base + tid < n) { offs[base + tid] = excl; cursor[base + tid] = excl; }
        __syncthreads();
        if (tid == 0) carry += buf[1023];
        __syncthreads();
    }
    if (tid == 0) offs[n] = carry;
}

__global__ void scatter_edges(const int* __restrict__ src, const int* __restrict__ dst,
                              int* __restrict__ cursor, int* __restrict__ col, int e) {
    int i = blockIdx.x * blockDim.x + threadIdx.x;
    if (i < e) {
        int d = dst[i];
        int pos = atomicAdd(&cursor[d], 1);
        col[pos] = src[i];
    }
}

// ---------------------------------------------------- sparse aggregation
// One wave per destination node. Each lane owns 6 contiguous floats of the
// 192-float row (lane*6 .. lane*6+5), loaded as 3x float2 (8B aligned) so the
// wave issues 3 coalesced 256B loads per edge and accumulates with pk_fma.
// x (38.4MB) is L2-resident on the 192MB L2. Manual 2-edge unroll doubles the
// number of independent L2 gathers in flight. Output permuted to [N][T][F].
__global__ __launch_bounds__(256) void aggregate(
    const float* __restrict__ x, const int* __restrict__ offs,
    const int* __restrict__ col, const float* __restrict__ dis,
    float* __restrict__ xagg)
{
    int wid  = (blockIdx.x * blockDim.x + threadIdx.x) >> 5;   // node
    int lane = threadIdx.x & 31;
    if (wid >= N_NODES) return;
    const int base = lane * 6;                  // 6 contiguous floats per lane
    float dr = dis[wid];
    float selfw = dr * dr;
    const v2f* xs = (const v2f*)(x + (size_t)wid * NODE_F + base);
    v2f a0 = selfw * xs[0];
    v2f a1 = selfw * xs[1];
    v2f a2 = selfw * xs[2];
    int kb = offs[wid], ke = offs[wid + 1];
    int k = kb;
    for (; k + 1 < ke; k += 2) {
        int s0 = col[k], s1 = col[k + 1];       // uniform across the wave
        float w0 = dis[s0] * dr;
        float w1 = dis[s1] * dr;
        const v2f* p0 = (const v2f*)(x + (size_t)s0 * NODE_F + base);
        const v2f* p1 = (const v2f*)(x + (size_t)s1 * NODE_F + base);
        v2f b00 = p0[0], b01 = p0[1], b02 = p0[2];
        v2f b10 = p1[0], b11 = p1[1], b12 = p1[2];
        a0 += w0 * b00; a1 += w0 * b01; a2 += w0 * b02;
        a0 += w1 * b10; a1 += w1 * b11; a2 += w1 * b12;
    }
    if (k < ke) {
        int s = col[k];
        float w = dis[s] * dr;
        const v2f* p = (const v2f*)(x + (size_t)s * NODE_F + base);
        a0 += w * p[0]; a1 += w * p[1]; a2 += w * p[2];
    }
    // write with [F][T] -> [T][F] permutation
    float av[6] = { a0[0], a0[1], a1[0], a1[1], a2[0], a2[1] };
    float* o = xagg + (size_t)wid * NODE_F;
#pragma unroll
    for (int j = 0; j < 6; j++) {
        int i = base + j;                      // in-node index, layout [F][T]
        int f = i / T_STEPS, t = i % T_STEPS;
        o[t * F_INF + f] = av[j];              // store as [T][F]
    }
}

// ---------------------------------------------------- weight fusion + swizzle
// Fused X-weights  Wf_g = Wg @ Lg_top   (16x32)
// H-weights        Lg_bot               (32x32)
// Both stored pre-swizzled into V_WMMA_F32_16X16X4_F32 B-fragment layout:
//   lane l holds B[K=4*ks+2*(l>>4) + {0,1}][N = nh*16 + (l&15)] as a float2.
__global__ void setup_weights(
    const float* Wz, const float* bz, const float* Wr, const float* br,
    const float* Wh, const float* bh,
    const float* LzW, const float* Lzb, const float* LrW, const float* Lrb,
    const float* LhW, const float* Lhb,
    const float* att,
    float* __restrict__ wsw, float* __restrict__ biasf,
    float* __restrict__ probs, float* __restrict__ pool)
{
    int tid = threadIdx.x;                       // 1 block, 256 threads
    const float* W[3]  = { Wz, Wr, Wh };
    const float* L[3]  = { LzW, LrW, LhW };
    const float* bg[3] = { bz, br, bh };
    const float* Lb[3] = { Lzb, Lrb, Lhb };

    // BX region: per gate 512 floats at wsw + g*1536
    for (int i = tid; i < 3 * 512; i += 256) {
        int g = i / 512, r = i % 512;
        int nh = r >> 8, r2 = r & 255;
        int ks = r2 >> 6, r3 = r2 & 63;
        int lane = r3 >> 1, p = r3 & 1;
        int k = 4 * ks + 2 * (lane >> 4) + p;          // input feature (0..15)
        int n = nh * 16 + (lane & 15);                 // output hid (0..31)
        float s = 0.f;
        for (int m = 0; m < HID; m++) s += W[g][k * HID + m] * L[g][m * HID + n];
        wsw[g * 1536 + r] = s;
    }
    // BH region: per gate 1024 floats at wsw + g*1536 + 512 (plain swizzle of L_bot)
    for (int i = tid; i < 3 * 1024; i += 256) {
        int g = i / 1024, r = i % 1024;
        int nh = r >> 9, r2 = r & 511;
        int ks = r2 >> 6, r3 = r2 & 63;
        int lane = r3 >> 1, p = r3 & 1;
        int kk = 4 * ks + 2 * (lane >> 4) + p;         // H dim (0..31)
        int n = nh * 16 + (lane & 15);
        wsw[g * 1536 + 512 + r] = L[g][(HID + kk) * HID + n];
    }
    // fused bias: bg @ L_top + Lgb
    for (int i = tid; i < 96; i += 256) {
        int g = i / 32, j = i % 32;
        float s = Lb[g][j];
        for (int k = 0; k < F_INF; k++) s += bg[g][k] * L[g][k * HID + j];
        biasf[i] = s;
    }
    if (tid < HID) pool[tid] = 0.f;
    if (tid == 0) {                                   // softmax(att), T=12
        float mx = att[0];
        for (int t = 1; t < T_STEPS; t++) mx = fmaxf(mx, att[t]);
        float e_[T_STEPS], se = 0.f;
        for (int t = 0; t < T_STEPS; t++) { e_[t] = __expf(att[t] - mx); se += e_[t]; }
        for (int t = 0; t < T_STEPS; t++) probs[t] = e_[t] / se;
    }
}

// ---------------------------------------------------- GRU dense phase (WMMA)
// One wave = one tile of 16 nodes. All GEMMs on v_wmma_f32_16x16x4_f32.
// H kept in C/D-layout registers for elementwise math; round-trips through a
// padded LDS buffer (stride 34 -> bank-conflict-free) to form A-fragments.
__global__ __launch_bounds__(256) void dense_gru(
    const float* __restrict__ xagg, const float* __restrict__ wsw,
    const float* __restrict__ biasf, const float* __restrict__ probs,
    float* __restrict__ pool, int nTiles)
{
    __shared__ __align__(16) float sH[8][16 * 34];
    __shared__ __align__(16) float sHR[8][16 * 34];
    const int lane = threadIdx.x & 31;
    const int wv   = threadIdx.x >> 5;
    int tile = blockIdx.x * 8 + wv;
    const bool valid = tile < nTiles;
    if (!valid) tile = nTiles - 1;      // clamp: duplicate work, guarded output,
                                        // keeps EXEC all-ones for WMMA + barriers
    const int node0 = tile * 16;
    const int mrow  = lane & 15;        // M row (node in tile) / N column
    const int kg    = lane >> 4;        // K-group (0 or 1)
    float* Hb  = sH[wv];
    float* HRb = sHR[wv];
    for (int i = lane; i < 16 * 34; i += 32) Hb[i] = 0.f;   // H0 = 0

    v8f Hlo = {0,0,0,0,0,0,0,0}, Hhi = Hlo;   // H in C layout (cols 0-15 / 16-31)
    v8f Alo = Hlo, Ahi = Hlo;                  // attention accumulator

    const v2f* BX[3]; const v2f* BH[3];
#pragma unroll
    for (int g = 0; g < 3; g++) {
        BX[g] = (const v2f*)(wsw + g * 1536);
        BH[g] = (const v2f*)(wsw + g * 1536 + 512);
    }
    float bcol[3][2];
#pragma unroll
    for (int g = 0; g < 3; g++)
#pragma unroll
        for (int h = 0; h < 2; h++) bcol[g][h] = biasf[g * 32 + h * 16 + mrow];

    const float* xbase = xagg + (size_t)(node0 + mrow) * NODE_F + 2 * kg;

    __syncthreads();

    for (int t = 0; t < T_STEPS; t++) {
        // A fragments of xagg_t (16x16, K = input feature)
        v2f ax[4];
        const float* xr = xbase + t * F_INF;
#pragma unroll
        for (int ks = 0; ks < 4; ks++) ax[ks] = *(const v2f*)(xr + 4 * ks);
        // A fragments of H (16x32, K = hid) from LDS
        v2f ah[8];
#pragma unroll
        for (int ks = 0; ks < 8; ks++)
            ah[ks] = *(const v2f*)(Hb + mrow * 34 + 4 * ks + 2 * kg);

        v8f Zlo, Zhi, Rlo, Rhi;
        {
            v8f out[2][2];
#pragma unroll
            for (int g = 0; g < 2; g++) {          // g=0: Z, g=1: R
#pragma unroll
                for (int h = 0; h < 2; h++) {
                    v8f c;
#pragma unroll
                    for (int i = 0; i < 8; i++) c[i] = bcol[g][h];
#pragma unroll
                    for (int ks = 0; ks < 4; ks++) {
                        v2f b = BX[g][(h * 4 + ks) * 32 + lane];
                        c = __builtin_amdgcn_wmma_f32_16x16x4_f32(
                                false, ax[ks], false, b, (short)0, c, false, false);
                    }
#pragma unroll
                    for (int ks = 0; ks < 8; ks++) {
                        v2f b = BH[g][(h * 8 + ks) * 32 + lane];
                        c = __builtin_amdgcn_wmma_f32_16x16x4_f32(
                                false, ah[ks], false, b, (short)0, c, false, false);
                    }
#pragma unroll
                    for (int i = 0; i < 8; i++)     // sigmoid
                        c[i] = 1.0f / (1.0f + __expf(-c[i]));
                    out[g][h] = c;
                }
            }
            Zlo = out[0][0]; Zhi = out[0][1];
            Rlo = out[1][0]; Rhi = out[1][1];
        }

        // H*R -> LDS (C layout scatter), then re-read as A fragments
        {
            v8f plo = Hlo * Rlo, phi = Hhi * Rhi;
#pragma unroll
            for (int v = 0; v < 8; v++) {
                HRb[(v + 8 * kg) * 34 + mrow]      = plo[v];
                HRb[(v + 8 * kg) * 34 + 16 + mrow] = phi[v];
            }
        }
        __syncthreads();

        v8f Tlo, Thi;
        {
            v2f ahr[8];
#pragma unroll
            for (int ks = 0; ks < 8; ks++)
                ahr[ks] = *(const v2f*)(HRb + mrow * 34 + 4 * ks + 2 * kg);
#pragma unroll
            for (int h = 0; h < 2; h++) {
                v8f c;
#pragma unroll
                for (int i = 0; i < 8; i++) c[i] = bcol[2][h];
#pragma unroll
                for (int ks = 0; ks < 4; ks++) {
                    v2f b = BX[2][(h * 4 + ks) * 32 + lane];
                    c = __builtin_amdgcn_wmma_f32_16x16x4_f32(
                            false, ax[ks], false, b, (short)0, c, false, false);
                }
#pragma unroll
                for (int ks = 0; ks < 8; ks++) {
                    v2f b = BH[2][(h * 8 + ks) * 32 + lane];
                    c = __builtin_amdgcn_wmma_f32_16x16x4_f32(
                            false, ahr[ks], false, b, (short)0, c, false, false);
                }
#pragma unroll
                for (int i = 0; i < 8; i++) {       // tanh
                    float e = __expf(2.0f * c[i]);
                    c[i] = 1.0f - 2.0f / (e + 1.0f);
                }
                if (h == 0) Tlo = c; else Thi = c;
            }
        }

        float p = probs[t];
        Hlo = Zlo * Hlo + (1.0f - Zlo) * Tlo;       // GRU update
        Hhi = Zhi * Hhi + (1.0f - Zhi) * Thi;
        Alo += p * Hlo;
        Ahi += p * Hhi;

#pragma unroll
        for (int v = 0; v < 8; v++) {               // H -> LDS for next step
            Hb[(v + 8 * kg) * 34 + mrow]      = Hlo[v];
            Hb[(v + 8 * kg) * 34 + 16 + mrow] = Hhi[v];
        }
        __syncthreads();
    }

    // relu + column sums into the global pool vector
    float slo = 0.f, shi = 0.f;
#pragma unroll
    for (int v = 0; v < 8; v++) {
        slo += fmaxf(Alo[v], 0.f);
        shi += fmaxf(Ahi[v], 0.f);
    }
    slo += __shfl_xor(slo, 16, 32);
    shi += __shfl_xor(shi, 16, 32);
    if (valid && kg == 0) {
        atomicAdd(&pool[mrow],      slo);
        atomicAdd(&pool[16 + mrow], shi);
    }
}

__global__ void finalize(const float* __restrict__ pool, const float* __restrict__ linW,
                         const float* __restrict__ linb, float* __restrict__ out) {
    int l = threadIdx.x;                       // 32 threads
    float v = (pool[l] / (float)N_NODES) * linW[l];
#pragma unroll
    for (int off = 16; off; off >>= 1) v += __shfl_xor(v, off, 32);
    if (l == 0) out[0] = fmaxf(v + linb[0], 0.f);
}

// ---------------------------------------------------------------- launcher
extern "C" void kernel_launch(void* const* d_in, const int* in_sizes, int n_in,
                              void* d_out, int out_size, void* d_ws, size_t ws_size,
                              hipStream_t stream) {
    (void)in_sizes; (void)n_in; (void)out_size; (void)ws_size;
    const float* x    = (const float*)d_in[0];
    const int*   ei   = (const int*)d_in[1];
    const float* att  = (const float*)d_in[2];
    const float* Wz   = (const float*)d_in[3];
    const float* bz   = (const float*)d_in[4];
    const float* Wr   = (const float*)d_in[5];
    const float* br   = (const float*)d_in[6];
    const float* Wh   = (const float*)d_in[7];
    const float* bh   = (const float*)d_in[8];
    const float* LzW  = (const float*)d_in[9];
    const float* Lzb  = (const float*)d_in[10];
    const float* LrW  = (const float*)d_in[11];
    const float* Lrb  = (const float*)d_in[12];
    const float* LhW  = (const float*)d_in[13];
    const float* Lhb  = (const float*)d_in[14];
    const float* linW = (const float*)d_in[15];
    const float* linb = (const float*)d_in[16];
    float* out = (float*)d_out;

    char* ws = (char*)d_ws;
    size_t off = 0;
    auto alloc = [&](size_t bytes) -> void* {
        off = (off + 255) & ~(size_t)255;
        void* p = ws + off;
        off += bytes;
        return p;
    };
    int*   indeg  = (int*)alloc((size_t)N_NODES * 4);
    int*   cursor = (int*)alloc((size_t)N_NODES * 4);
    int*   offs   = (int*)alloc((size_t)(N_NODES + 1) * 4);
    int*   col    = (int*)alloc((size_t)N_EDGES * 4);
    float* dis    = (float*)alloc((size_t)N_NODES * 4);
    float* xagg   = (float*)alloc((size_t)N_NODES * NODE_F * 4);
    float* wsw    = (float*)alloc(4608 * 4);
    float* biasf  = (float*)alloc(96 * 4);
    float* probs  = (float*)alloc(T_STEPS * 4);
    float* pool   = (float*)alloc(HID * 4);

    const int* src = ei;
    const int* dst = ei + N_EDGES;

    zero_i32     <<<(N_NODES + 255) / 256, 256, 0, stream>>>(indeg, N_NODES);
    count_indeg  <<<(N_EDGES + 255) / 256, 256, 0, stream>>>(dst, indeg, N_EDGES);
    compute_dis  <<<(N_NODES + 255) / 256, 256, 0, stream>>>(indeg, dis, N_NODES);
    scan_offsets <<<1, 1024, 0, stream>>>(indeg, offs, cursor, N_NODES);
    scatter_edges<<<(N_EDGES + 255) / 256, 256, 0, stream>>>(src, dst, cursor, col, N_EDGES);
    setup_weights<<<1, 256, 0, stream>>>(Wz, bz, Wr, br, Wh, bh,
                                         LzW, Lzb, LrW, Lrb, LhW, Lhb,
                                         att, wsw, biasf, probs, pool);
    aggregate    <<<(N_NODES * 32 + 255) / 256, 256, 0, stream>>>(x, offs, col, dis, xagg);

    int nTiles = N_NODES / 16;                       // 3125
    dense_gru    <<<(nTiles + 7) / 8, 256, 0, stream>>>(xagg, wsw, biasf, probs, pool, nTiles);
    finalize     <<<1, 32, 0, stream>>>(pool, linW, linb, out);
}